// MSMESAGE_3453153706306
// MI455X (gfx1250) — compile-verified
//
#include <hip/hip_runtime.h>
#include <hip/hip_bf16.h>

// ---- problem constants (match reference) ----
#define GG    64
#define NN_   512
#define HH    64
#define NTT   32768            // GG*NN_
#define DEGC  32
#define ETT   (NTT*DEGC)       // 1048576
#define KPP   359
#define NPP   (GG*KPP)         // 22976
#define FIN   128
#define NCC   2

typedef __attribute__((ext_vector_type(2))) float v2f;
typedef __attribute__((ext_vector_type(8))) float v8f;

// ---------------- fills ----------------
__global__ void k_fill_f(float* p, float v, int n) {
  int i = blockIdx.x * blockDim.x + threadIdx.x;
  if (i < n) p[i] = v;
}
__global__ void k_fill_i(int* p, int v, int n) {
  int i = blockIdx.x * blockDim.x + threadIdx.x;
  if (i < n) p[i] = v;
}

// ---------------- e = coords @ pos_W + pos_b  ([NT,3]@[3,64]) ----------------
__global__ void k_pos(const float* __restrict__ coords, const float* __restrict__ W,
                      const float* __restrict__ b, float* __restrict__ e) {
  int i = blockIdx.x * blockDim.x + threadIdx.x;   // NTT*64 threads
  int n = i >> 6, f = i & 63;
  float c0 = coords[n*3+0], c1 = coords[n*3+1], c2 = coords[n*3+2];
  e[i] = c0*W[f] + c1*W[64+f] + c2*W[128+f] + b[f];
}

// ---------------- Ae[row] += edge_attr * e[col]  (stage 1) ----------------
__global__ void k_scatter_ae(const int* __restrict__ row, const int* __restrict__ col,
                             const float* __restrict__ ea, const float* __restrict__ e,
                             float* __restrict__ Ae) {
  int t = blockIdx.x * blockDim.x + threadIdx.x;   // ETT*16 threads
  int ed = t >> 4, fo = (t & 15) << 2;
  int r = row[ed], c = col[ed];
  float w = ea[ed];
  const float* src = e + (size_t)c*64 + fo;
  float* dst = Ae + (size_t)r*64 + fo;
#pragma unroll
  for (int i = 0; i < 4; ++i) atomicAdd(dst + i, w * src[i]);
}

// ---------------- agg[col] += x[row]; deg[col] += 1  (SAGE-1 mean aggr) ----------------
__global__ void k_scatter_agg1(const int* __restrict__ row, const int* __restrict__ col,
                               const float* __restrict__ x, float* __restrict__ agg,
                               float* __restrict__ deg) {
  int t = blockIdx.x * blockDim.x + threadIdx.x;   // ETT*32 threads
  int ed = t >> 5, fo = (t & 31) << 2;
  int r = row[ed], c = col[ed];
  const float* src = x + (size_t)r*128 + fo;
  float* dst = agg + (size_t)c*128 + fo;
#pragma unroll
  for (int i = 0; i < 4; ++i) atomicAdd(dst + i, src[i]);
  if (fo == 0) atomicAdd(deg + c, 1.0f);
}

// ---------------- WMMA fp32 GEMM: out[M,64] = A1@W1 + (A2/deg)@W2 + bias ----------------
// block = 128 threads = 4 waves; each wave owns one 16-col tile, block owns 16 rows.
// V_WMMA_F32_16X16X4_F32 layouts per ISA: A 16x4 (2 VGPR), B 4x16 (2 VGPR), C 16x16 (8 VGPR).
__global__ void k_gemm64(const float* __restrict__ A1, const float* __restrict__ W1, int K1,
                         const float* __restrict__ A2, const float* __restrict__ W2, int K2,
                         const float* __restrict__ deg, const float* __restrict__ bias,
                         float* __restrict__ out) {
  int mbase = blockIdx.x << 4;
  int lane  = threadIdx.x & 31;
  int nbase = (threadIdx.x >> 5) << 4;
  int l15   = lane & 15;
  int hi    = lane >> 4;            // 0: K=0,1 / rows 0..7 ; 1: K=2,3 / rows 8..15
  int mrow  = mbase + l15;
  int ncol  = nbase + l15;
  int koff  = hi << 1;
  v8f acc = {};
  const float* arow = A1 + (size_t)mrow * K1;
  __builtin_prefetch(arow, 0, 3);   // global_prefetch_b8
  for (int k = 0; k < K1; k += 4) {
    v2f a, b;
    a.x = arow[k + koff];            a.y = arow[k + koff + 1];
    b.x = W1[(k + koff) * 64 + ncol]; b.y = W1[(k + koff + 1) * 64 + ncol];
    acc = __builtin_amdgcn_wmma_f32_16x16x4_f32(false, a, false, b, (short)0, acc,
                                                false, false);
  }
  if (K2 > 0) {
    float invd = 1.0f / fmaxf(deg[mrow], 1.0f);
    const float* arow2 = A2 + (size_t)mrow * K2;
    for (int k = 0; k < K2; k += 4) {
      v2f a, b;
      a.x = arow2[k + koff] * invd;     a.y = arow2[k + koff + 1] * invd;
      b.x = W2[(k + koff) * 64 + ncol]; b.y = W2[(k + koff + 1) * 64 + ncol];
      acc = __builtin_amdgcn_wmma_f32_16x16x4_f32(false, a, false, b, (short)0, acc,
                                                  false, false);
    }
  }
  int crow = mbase + (hi << 3);
  float bb = bias[ncol];
#pragma unroll
  for (int r = 0; r < 8; ++r)
    out[(size_t)(crow + r) * 64 + ncol] = acc[r] + bb;
}

// ---------------- delta[n,h] = sum_d Ae[n,d] * Bk[class][h,d] ----------------
__global__ void k_delta(const float* __restrict__ Ae, const int* __restrict__ cls,
                        const int* __restrict__ perm, const float* __restrict__ Bk,
                        float* __restrict__ delta) {
  int n = blockIdx.x, t = threadIdx.x;      // 64 threads
  __shared__ float a[64];
  a[t] = Ae[(size_t)n*64 + t];
  __syncthreads();
  int node = perm ? perm[n] : n;
  const float* B = Bk + (size_t)cls[node] * 4096 + (size_t)t * 64;
  float s = 0.f;
#pragma unroll 8
  for (int d = 0; d < 64; ++d) s += a[d] * B[d];
  delta[(size_t)n*64 + t] = s;
}

// ---------------- out = (Bk + delta⊗1) @ (hin + delta) = Bk·hh + delta·Σhh ----------------
__global__ void k_prop(const float* __restrict__ hin, const float* __restrict__ delta,
                       const int* __restrict__ cls, const int* __restrict__ perm,
                       const float* __restrict__ Bk, float* __restrict__ out) {
  int n = blockIdx.x, t = threadIdx.x;      // 64 threads
  __shared__ float hh[64];
  __shared__ float red[64];
  float dl = delta[(size_t)n*64 + t];
  float v  = hin[(size_t)n*64 + t] + dl;
  hh[t] = v; red[t] = v;
  __syncthreads();
  for (int o = 32; o > 0; o >>= 1) { if (t < o) red[t] += red[t + o]; __syncthreads(); }
  float s = red[0];
  int node = perm ? perm[n] : n;
  const float* B = Bk + (size_t)cls[node] * 4096 + (size_t)t * 64;
  float acc = 0.f;
#pragma unroll 8
  for (int j = 0; j < 64; ++j) acc += B[j] * hh[j];
  out[(size_t)n*64 + t] = acc + dl * s;
}

// ---------------- graph_norm + ELU (graphs are contiguous, nper nodes each) ----------------
__global__ void k_gn_elu(const float* __restrict__ hin, float* __restrict__ out,
                         const float* __restrict__ w, const float* __restrict__ b,
                         const float* __restrict__ ms, int nper) {
  int g = blockIdx.x, f = threadIdx.x;      // 64 threads
  const float* base = hin + (size_t)g * nper * 64;
  float* obase = out + (size_t)g * nper * 64;
  float mean = 0.f;
  for (int n = 0; n < nper; ++n) mean += base[(size_t)n*64 + f];
  mean /= (float)nper;
  float m = ms[f] * mean;
  float var = 0.f;
  for (int n = 0; n < nper; ++n) { float d = base[(size_t)n*64 + f] - m; var += d * d; }
  var /= (float)nper;
  float inv = 1.0f / sqrtf(var + 1e-5f);
  float wf = w[f], bf = b[f];
  for (int n = 0; n < nper; ++n) {
    float d = base[(size_t)n*64 + f] - m;
    float y = wf * d * inv + bf;
    obase[(size_t)n*64 + f] = (y > 0.f) ? y : (expf(y) - 1.f);
  }
}

// ---------------- score[n] = (h[n]·p)/||p|| ----------------
__global__ void k_score(const float* __restrict__ h, const float* __restrict__ p,
                        float* __restrict__ score) {
  int n = blockIdx.x * blockDim.x + threadIdx.x;
  if (n >= NTT) return;
  float nn = 0.f, d = 0.f;
  const float* hr = h + (size_t)n * 64;
  for (int f = 0; f < 64; ++f) { float pv = p[f]; nn += pv * pv; d += hr[f] * pv; }
  score[n] = d / sqrtf(nn);
}

// ---------------- top-KP per graph by rank counting (matches top_k ordering) ----------------
__global__ void k_topk(const float* __restrict__ score, const float* __restrict__ h,
                       int* __restrict__ perm, int* __restrict__ nodemap,
                       float* __restrict__ h2) {
  int g = blockIdx.x, t = threadIdx.x;      // 512 threads
  __shared__ float s[512];
  int node = g * 512 + t;
  float sv = score[node];
  s[t] = sv;
  __syncthreads();
  int rank = 0;
  for (int j = 0; j < 512; ++j) {
    float o = s[j];
    rank += (o > sv) || (o == sv && j < t);   // unique ranks; ties by lower index first
  }
  if (rank < KPP) {
    int slot = g * KPP + rank;
    perm[slot] = node;
    nodemap[node] = slot;
    float gate = tanhf(sv);
    const float* hr = h + (size_t)node * 64;
    float* dst = h2 + (size_t)slot * 64;
    for (int f = 0; f < 64; ++f) dst[f] = hr[f] * gate;
  }
}

// ---------------- stage-2 pre-propagation scatter: Ae2[s2] += e[col]; deg2[t2]+=1 ----------------
__global__ void k_scatter2a(const int* __restrict__ row, const int* __restrict__ col,
                            const int* __restrict__ nodemap, const float* __restrict__ e,
                            float* __restrict__ Ae2, float* __restrict__ deg2) {
  int t = blockIdx.x * blockDim.x + threadIdx.x;   // ETT*16 threads
  int ed = t >> 4, fo = (t & 15) << 2;
  int s2 = nodemap[row[ed]], t2 = nodemap[col[ed]];
  if (s2 < 0 || t2 < 0) return;
  const float* src = e + (size_t)col[ed]*64 + fo;  // e2[t2] == e[original col]
  float* dst = Ae2 + (size_t)s2*64 + fo;
#pragma unroll
  for (int i = 0; i < 4; ++i) atomicAdd(dst + i, src[i]);
  if (fo == 0) atomicAdd(deg2 + t2, 1.0f);
}

// ---------------- stage-2 post-propagation scatter: agg2[t2] += h2q[s2] ----------------
__global__ void k_scatter2b(const int* __restrict__ row, const int* __restrict__ col,
                            const int* __restrict__ nodemap, const float* __restrict__ h2q,
                            float* __restrict__ agg2) {
  int t = blockIdx.x * blockDim.x + threadIdx.x;   // ETT*16 threads
  int ed = t >> 4, fo = (t & 15) << 2;
  int s2 = nodemap[row[ed]], t2 = nodemap[col[ed]];
  if (s2 < 0 || t2 < 0) return;
  const float* src = h2q + (size_t)s2*64 + fo;
  float* dst = agg2 + (size_t)t2*64 + fo;
#pragma unroll
  for (int i = 0; i < 4; ++i) atomicAdd(dst + i, src[i]);
}

// ---------------- readout: [mean | max | sum] per graph -> gmat[G,192] ----------------
__global__ void k_readout(const float* __restrict__ h, float* __restrict__ gmat) {
  int g = blockIdx.x, f = threadIdx.x;      // 64 threads
  const float* base = h + (size_t)g * KPP * 64;
  float s = 0.f, mx = -3.402823466e38f;
  for (int n = 0; n < KPP; ++n) {
    float v = base[(size_t)n*64 + f];
    s += v; mx = fmaxf(mx, v);
  }
  gmat[(size_t)g*192 + f]       = s / (float)KPP;
  gmat[(size_t)g*192 + 64 + f]  = mx;
  gmat[(size_t)g*192 + 128 + f] = s;
}

// ---------------- head: relu(gl1) @ lin2 + b -> log_softmax ----------------
__global__ void k_head(const float* __restrict__ gl1, const float* __restrict__ W2,
                       const float* __restrict__ b2, float* __restrict__ out) {
  int g = blockIdx.x, t = threadIdx.x;      // 64 threads
  __shared__ float r[64];
  __shared__ float z[2];
  float v = gl1[(size_t)g*64 + t];
  r[t] = (v > 0.f) ? v : 0.f;
  __syncthreads();
  if (t < 2) {
    float a = b2[t];
    for (int j = 0; j < 64; ++j) a += r[j] * W2[j*2 + t];
    z[t] = a;
  }
  __syncthreads();
  if (t == 0) {
    float m = fmaxf(z[0], z[1]);
    float lse = m + logf(expf(z[0] - m) + expf(z[1] - m));
    out[g*2 + 0] = z[0] - lse;
    out[g*2 + 1] = z[1] - lse;
  }
}

// =====================================================================
extern "C" void kernel_launch(void* const* d_in, const int* in_sizes, int n_in,
                              void* d_out, int out_size, void* d_ws, size_t ws_size,
                              hipStream_t stream) {
  const float* x      = (const float*)d_in[0];
  const int*   eidx   = (const int*)  d_in[1];
  const int*   row    = eidx;
  const int*   col    = eidx + ETT;
  const float* eattr  = (const float*)d_in[2];
  const float* coords = (const float*)d_in[3];
  const int*   cls    = (const int*)  d_in[4];
  const float* Bk     = (const float*)d_in[5];
  const float* posW   = (const float*)d_in[6];
  const float* posb   = (const float*)d_in[7];
  const float* W1s    = (const float*)d_in[8];
  const float* W1n    = (const float*)d_in[9];
  const float* b1     = (const float*)d_in[10];
  const float* gn1w   = (const float*)d_in[11];
  const float* gn1b   = (const float*)d_in[12];
  const float* gn1ms  = (const float*)d_in[13];
  const float* poolp  = (const float*)d_in[14];
  const float* W2s    = (const float*)d_in[15];
  const float* W2n    = (const float*)d_in[16];
  const float* b2     = (const float*)d_in[17];
  const float* gn2w   = (const float*)d_in[18];
  const float* gn2b   = (const float*)d_in[19];
  const float* gn2ms  = (const float*)d_in[20];
  const float* lin1W  = (const float*)d_in[21];
  const float* lin1b  = (const float*)d_in[22];
  const float* lin2W  = (const float*)d_in[23];
  const float* lin2b  = (const float*)d_in[24];
  float* out = (float*)d_out;

  float* ws = (float*)d_ws;
  size_t o = 0;
  float* e      = ws + o; o += (size_t)NTT*64;
  float* h      = ws + o; o += (size_t)NTT*64;
  float* delta  = ws + o; o += (size_t)NTT*64;
  float* tmp    = ws + o; o += (size_t)NTT*64;
  float* Ae     = ws + o; o += (size_t)NTT*64;
  float* agg1   = ws + o; o += (size_t)NTT*128;
  float* deg1   = ws + o; o += (size_t)NTT;
  float* score  = ws + o; o += (size_t)NTT;
  float* h2     = ws + o; o += (size_t)NPP*64;
  float* Ae2    = ws + o; o += (size_t)NPP*64;
  float* agg2   = ws + o; o += (size_t)NPP*64;
  float* delta2 = ws + o; o += (size_t)NPP*64;
  float* h2q    = ws + o; o += (size_t)NPP*64;
  float* h3     = ws + o; o += (size_t)NPP*64;
  float* h3n    = ws + o; o += (size_t)NPP*64;
  float* deg2   = ws + o; o += (size_t)NPP;
  float* gmat   = ws + o; o += (size_t)GG*192;
  float* gl1    = ws + o; o += (size_t)GG*64;
  int* perm     = (int*)(ws + o); o += (size_t)NPP;
  int* nodemap  = (int*)(ws + o); o += (size_t)NTT;

  auto blks = [](long n, int b) { return dim3((unsigned)((n + b - 1) / b)); };

  // --- zero accumulators / init node map ---
  k_fill_f<<<blks((long)NTT*64, 256), 256, 0, stream>>>(Ae, 0.f, NTT*64);
  k_fill_f<<<blks((long)NTT*128, 256), 256, 0, stream>>>(agg1, 0.f, NTT*128);
  k_fill_f<<<blks(NTT, 256), 256, 0, stream>>>(deg1, 0.f, NTT);
  k_fill_f<<<blks((long)NPP*64, 256), 256, 0, stream>>>(Ae2, 0.f, NPP*64);
  k_fill_f<<<blks((long)NPP*64, 256), 256, 0, stream>>>(agg2, 0.f, NPP*64);
  k_fill_f<<<blks(NPP, 256), 256, 0, stream>>>(deg2, 0.f, NPP);
  k_fill_i<<<blks(NTT, 256), 256, 0, stream>>>(nodemap, -1, NTT);

  // --- stage 1 ---
  k_pos<<<blks((long)NTT*64, 256), 256, 0, stream>>>(coords, posW, posb, e);
  k_scatter_ae<<<blks((long)ETT*16, 256), 256, 0, stream>>>(row, col, eattr, e, Ae);
  k_scatter_agg1<<<blks((long)ETT*32, 256), 256, 0, stream>>>(row, col, x, agg1, deg1);
  k_gemm64<<<NTT/16, 128, 0, stream>>>(x, W1s, FIN, agg1, W1n, FIN, deg1, b1, h);
  k_delta<<<NTT, 64, 0, stream>>>(Ae, cls, nullptr, Bk, delta);
  k_gn_elu<<<GG, 64, 0, stream>>>(h, tmp, gn1w, gn1b, gn1ms, NN_);
  k_prop<<<NTT, 64, 0, stream>>>(tmp, delta, cls, nullptr, Bk, h);

  // --- top-k pooling ---
  k_score<<<blks(NTT, 256), 256, 0, stream>>>(h, poolp, score);
  k_topk<<<GG, 512, 0, stream>>>(score, h, perm, nodemap, h2);

  // --- stage 2 ---
  k_scatter2a<<<blks((long)ETT*16, 256), 256, 0, stream>>>(row, col, nodemap, e, Ae2, deg2);
  k_delta<<<NPP, 64, 0, stream>>>(Ae2, cls, perm, Bk, delta2);
  k_prop<<<NPP, 64, 0, stream>>>(h2, delta2, cls, perm, Bk, h2q);
  k_scatter2b<<<blks((long)ETT*16, 256), 256, 0, stream>>>(row, col, nodemap, h2q, agg2);
  k_gemm64<<<NPP/16, 128, 0, stream>>>(h2q, W2s, HH, agg2, W2n, HH, deg2, b2, h3);
  k_gn_elu<<<GG, 64, 0, stream>>>(h3, h3n, gn2w, gn2b, gn2ms, KPP);

  // --- readout ---
  k_readout<<<GG, 64, 0, stream>>>(h3n, gmat);
  k_gemm64<<<GG/16, 128, 0, stream>>>(gmat, lin1W, 192, nullptr, nullptr, 0, nullptr,
                                      lin1b, gl1);
  k_head<<<GG, 64, 0, stream>>>(gl1, lin2W, lin2b, out);
}